// DLI_loss_3_6614249636367
// MI455X (gfx1250) — compile-verified
//
#include <hip/hip_runtime.h>
#include <hip/hip_bf16.h>
#include <math.h>

typedef __attribute__((ext_vector_type(2))) float v2f;
typedef __attribute__((ext_vector_type(8))) float v8f;

#define BSZ 128
#define T   512
#define ENC 128
#define HID 128

// ---------------------------------------------------------------------------
// The loss algebra: logits[b,j,k] = a[b,j] + cterm[b,k] + fc_b, and both the
// logsumexp and the positive term share the per-(b,j) constant a[b,j]+fc_b,
// so it cancels: loss_pair(b, j=k-3) = LSE_{k'=k..len-1}(cterm[b,k']) - cterm[b,k].
// The LSTM never reaches the output. Only cterm = (enc ⊙ mask) @ fc_w[H:] and a
// per-b suffix-LSE scan are needed.  Since cterm[b,k] is only consumed for
// k < len_b (where mask == 1), the mask multiply is dead work too.
// ---------------------------------------------------------------------------

// ---------------------------------------------------------------------------
// Kernel A: cterm[m] = sum_k enc[m,k] * fc_w[HID+k],  m = b*T + t
// GEMV on the fp32 WMMA unit: 32 x V_WMMA_F32_16X16X4_F32 per wave over K=128.
// B column 0 carries the weight vector; one wave per 16-row tile; 65536 rows
// = 4096 tiles exactly, so EXEC stays all-ones for every WMMA.
// Fragment layouts (ISA 7.12.2):
//   A 16x4 f32: lanes 0-15 -> M=lane, K={0,1}; lanes 16-31 -> M=lane-16, K={2,3}
//   B 4x16 f32: lanes 0-15 -> N=lane, K={0,1}; lanes 16-31 -> N=lane-16, K={2,3}
//   D 16x16:    VGPR r: lanes 0-15 -> M=r; lanes 16-31 -> M=8+r; N=lane%16
// ---------------------------------------------------------------------------
__global__ void cterm_wmma_kernel(const float* __restrict__ enc,
                                  const float* __restrict__ fc_w,   // [HID+ENC]
                                  float*       __restrict__ cterm)  // [BSZ*T]
{
  const int wave = (blockIdx.x * blockDim.x + threadIdx.x) >> 5;
  const int lane = threadIdx.x & 31;
  const int m0   = wave * 16;
  if (m0 >= BSZ * T) return;               // grid sized exactly; never taken

  const int half = lane >> 4;              // 0: lanes 0-15, 1: lanes 16-31
  const int l16  = lane & 15;
  const float* __restrict__ arow = enc  + (size_t)(m0 + l16) * ENC;
  const float* __restrict__ wcol = fc_w + HID;   // w[H:]

  v8f acc = (v8f){0.f, 0.f, 0.f, 0.f, 0.f, 0.f, 0.f, 0.f};

  #pragma unroll
  for (int kk = 0; kk < ENC / 4; ++kk) {
    const int k0 = kk * 4 + half * 2;      // this lane's K pair within the step
    v2f a  = *(const v2f*)(arow + k0);     // global_load_b64: the real traffic
    v2f wv = *(const v2f*)(wcol + k0);     // tiny, L0-resident
    v2f bf;
    bf.x = (l16 == 0) ? wv.x : 0.f;        // B column 0 = weight vector
    bf.y = (l16 == 0) ? wv.y : 0.f;
    acc = __builtin_amdgcn_wmma_f32_16x16x4_f32(
        false, a, false, bf, (short)0, acc, false, false);
  }

  // Extract D column 0: lane 0 -> rows m0..m0+7, lane 16 -> rows m0+8..m0+15.
  if (l16 == 0) {
    #pragma unroll
    for (int r = 0; r < 8; ++r)
      cterm[m0 + half * 8 + r] = acc[r];
  }
}

// ---------------------------------------------------------------------------
// Kernel B: one block per batch row b (128 blocks x 128 threads, 4 k's/thread).
// Two-pass suffix logsumexp with a single global max Mg over k in [3, len):
//   S[k] = sum_{k'>=k, k'<len} exp(c[k'] - Mg)   (parallel suffix sum)
//   loss += log(S[k]) + Mg - c[k]                (for each k in [3, len))
// which equals LSE(suffix from k) - c[k] exactly in real arithmetic.
// ---------------------------------------------------------------------------
#define TB 128
__global__ void suffix_lse_kernel(const int*   __restrict__ mask,
                                  const float* __restrict__ cterm,
                                  float*       __restrict__ bloss,  // [BSZ]
                                  float*       __restrict__ bcnt)   // [BSZ]
{
  const int b   = blockIdx.x;
  const int tid = threadIdx.x;
  __shared__ float sred[TB];
  __shared__ int   slen;

  // length = sum(mask[b,:])
  const int* mrow = mask + b * T;
  int lp = 0;
  #pragma unroll
  for (int j = 0; j < 4; ++j) lp += mrow[tid * 4 + j];
  sred[tid] = (float)lp;
  __syncthreads();
  for (int off = TB / 2; off > 0; off >>= 1) {
    if (tid < off) sred[tid] += sred[tid + off];
    __syncthreads();
  }
  if (tid == 0) slen = (int)sred[0];
  __syncthreads();
  const int len = slen;
  __syncthreads();

  // load chunk + validity
  float c4[4]; bool v4[4];
  const float* crow = cterm + b * T;
  #pragma unroll
  for (int j = 0; j < 4; ++j) {
    const int k = tid * 4 + j;
    c4[j] = crow[k];
    v4[j] = (k >= 3) && (k < len);
  }

  // global max over valid range
  float mloc = -INFINITY;
  #pragma unroll
  for (int j = 0; j < 4; ++j) if (v4[j]) mloc = fmaxf(mloc, c4[j]);
  sred[tid] = mloc;
  __syncthreads();
  for (int off = TB / 2; off > 0; off >>= 1) {
    if (tid < off) sred[tid] = fmaxf(sred[tid], sred[tid + off]);
    __syncthreads();
  }
  const float Mg = sred[0];
  __syncthreads();

  // exponentials + chunk total
  float e4[4], tsum = 0.f;
  #pragma unroll
  for (int j = 0; j < 4; ++j) {
    e4[j] = v4[j] ? expf(c4[j] - Mg) : 0.f;
    tsum += e4[j];
  }

  // inclusive suffix scan of chunk totals across threads (Hillis-Steele)
  sred[tid] = tsum;
  __syncthreads();
  for (int off = 1; off < TB; off <<= 1) {
    const float v = (tid + off < TB) ? sred[tid + off] : 0.f;
    __syncthreads();
    sred[tid] += v;
    __syncthreads();
  }
  const float excl = (tid + 1 < TB) ? sred[tid + 1] : 0.f;  // strictly-after sum
  __syncthreads();

  // within-chunk suffix + loss contributions
  float run = excl, loss = 0.f;
  #pragma unroll
  for (int j = 3; j >= 0; --j) {
    run += e4[j];
    if (v4[j]) loss += logf(run) + Mg - c4[j];
  }

  // block-reduce loss
  sred[tid] = loss;
  __syncthreads();
  for (int off = TB / 2; off > 0; off >>= 1) {
    if (tid < off) sred[tid] += sred[tid + off];
    __syncthreads();
  }
  if (tid == 0) { bloss[b] = sred[0]; bcnt[b] = (float)(len - 3); }
}

// ---------------------------------------------------------------------------
// Kernel C: deterministic final reduction, out = sum(bloss) / sum(bcnt)
// ---------------------------------------------------------------------------
__global__ void finalize_kernel(const float* __restrict__ bloss,
                                const float* __restrict__ bcnt,
                                float*       __restrict__ out)
{
  __shared__ float sl[BSZ], sc[BSZ];
  const int t = threadIdx.x;
  sl[t] = bloss[t];
  sc[t] = bcnt[t];
  __syncthreads();
  for (int off = BSZ / 2; off > 0; off >>= 1) {
    if (t < off) { sl[t] += sl[t + off]; sc[t] += sc[t + off]; }
    __syncthreads();
  }
  if (t == 0) out[0] = sl[0] / sc[0];
}

// ---------------------------------------------------------------------------
// Inputs (setup_inputs order):
//   0 encoder_output [128,512,128] f32   1 mask [128,512] i32
//   2 W_ih  3 W_hh  4 b_ih  5 b_hh  7 fc_b   (algebraically cancel)
//   6 fc_w [1,256] f32
// Output: scalar f32 loss.
// ---------------------------------------------------------------------------
extern "C" void kernel_launch(void* const* d_in, const int* in_sizes, int n_in,
                              void* d_out, int out_size, void* d_ws, size_t ws_size,
                              hipStream_t stream) {
  const float* enc  = (const float*)d_in[0];
  const int*   mask = (const int*)  d_in[1];
  const float* fc_w = (const float*)d_in[6];
  float*       out  = (float*)d_out;

  float* cterm = (float*)d_ws;                 // BSZ*T floats = 256 KB
  float* bloss = cterm + BSZ * T;              // [BSZ]
  float* bcnt  = bloss + BSZ;                  // [BSZ]

  // Kernel A: 65536 rows / 16 rows-per-wave = 4096 waves; 8 waves per block.
  const int threads = 256;
  const int blocks  = ((BSZ * T) / 16 * 32) / threads;   // 512
  cterm_wmma_kernel<<<blocks, threads, 0, stream>>>(enc, fc_w, cterm);

  suffix_lse_kernel<<<BSZ, TB, 0, stream>>>(mask, cterm, bloss, bcnt);
  finalize_kernel<<<1, BSZ, 0, stream>>>(bloss, bcnt, out);
}